// SplitLinear_87454124081203
// MI455X (gfx1250) — compile-verified
//
#include <hip/hip_runtime.h>
#include <stdint.h>

// SplitLinear: y[t,g] = sum_h x[t, g*5+h] * w[g,h] + bias[g]
// T=512, G=30490, H=5. ~150x memory bound (375 MB traffic, 0.156 GFLOP) ->
// stream x through LDS with CDNA5 async global->LDS copies (ASYNCcnt path),
// NT temporal hints for the single-use stream, scalar FMA compute.

#define G_TOTAL 30490
#define HID 5
#define T_TOTAL 512
#define GC 256                         // groups per block
#define TC 8                           // time rows per block
#define ROW_FLOATS (GC * HID)          // 1280 floats per staged row
#define ROW_BYTES (ROW_FLOATS * 4)     // 5120 bytes per staged row (8B aligned chunks)
#define CHUNKS (ROW_BYTES / 256)       // 20 async b64 instructions per row (32 lanes * 8B)

__global__ __launch_bounds__(256) void splitlinear_async_kernel(
    const float* __restrict__ x, const float* __restrict__ weight,
    const float* __restrict__ bias, float* __restrict__ y) {
  __shared__ float smem[TC * ROW_FLOATS];  // 40 KB LDS tile of x

  const int tid = threadIdx.x;
  const int lane = tid & 31;
  const int wave = tid >> 5;  // 0..7, one staged row per wave
  const int g0 = blockIdx.x * GC;
  const int t0 = blockIdx.y * TC;

  // ---------------- async stage: global x tile -> LDS ----------------
  const uint64_t xbase = (uint64_t)(uintptr_t)x;
  // clamp limit keeps tail-chunk reads inside the allocation (8B aligned)
  const uint64_t xlimit =
      xbase + (uint64_t)T_TOTAL * (uint64_t)(G_TOTAL * HID) * 4ull - 8ull;
  // low 32 bits of a flat shared pointer == wave-relative LDS byte offset
  const uint32_t smem_base = (uint32_t)(uintptr_t)(&smem[0]);

  {
    const int r = wave;  // TC==8 rows, 8 waves: one row per wave
    const uint64_t rowbase =
        xbase +
        ((uint64_t)(t0 + r) * (uint64_t)(G_TOTAL * HID) + (uint64_t)g0 * HID) *
            4ull;
    const uint32_t lds_row = smem_base + (uint32_t)(r * ROW_BYTES);
#pragma unroll
    for (int c = 0; c < CHUNKS; ++c) {
      const uint32_t off = (uint32_t)(c * 256 + lane * 8);
      uint64_t ga = rowbase + off;
      if (ga > xlimit) ga = xlimit;  // stay in-bounds on the final tile
      const uint32_t la = lds_row + off;
      // NT hint: x is a 312MB single-use stream (> 192MB L2) -> don't cache
      asm volatile("global_load_async_to_lds_b64 %0, %1, off th:TH_LOAD_NT" ::
                       "v"(la),
                   "v"(ga)
                   : "memory");
    }
  }

  // per-thread weights/bias (overlaps with in-flight async copies);
  // these ARE reused across the 64 t-chunks -> default (cached) policy
  const int g = g0 + tid;
  float w0 = 0.f, w1 = 0.f, w2 = 0.f, w3 = 0.f, w4 = 0.f, b = 0.f;
  if (g < G_TOTAL) {
    const float* wp = weight + (size_t)g * HID;
    w0 = wp[0];
    w1 = wp[1];
    w2 = wp[2];
    w3 = wp[3];
    w4 = wp[4];
    b = bias[g];
  }

  // drain this wave's async copies, then block-wide barrier so every wave
  // sees every staged row
  asm volatile("s_wait_asynccnt 0" ::: "memory");
  __syncthreads();

  // ---------------- compute: 5-FMA dot per output ----------------
  if (g < G_TOTAL) {
#pragma unroll
    for (int r = 0; r < TC; ++r) {
      const float* xr = &smem[r * ROW_FLOATS + tid * HID];
      float acc = fmaf(xr[0], w0, b);
      acc = fmaf(xr[1], w1, acc);
      acc = fmaf(xr[2], w2, acc);
      acc = fmaf(xr[3], w3, acc);
      acc = fmaf(xr[4], w4, acc);
      // y is write-once, never re-read by this dispatch -> NT store
      __builtin_nontemporal_store(acc, &y[(size_t)(t0 + r) * G_TOTAL + g]);
    }
  }
}

extern "C" void kernel_launch(void* const* d_in, const int* in_sizes, int n_in,
                              void* d_out, int out_size, void* d_ws,
                              size_t ws_size, hipStream_t stream) {
  (void)in_sizes;
  (void)n_in;
  (void)out_size;
  (void)d_ws;
  (void)ws_size;
  const float* x = (const float*)d_in[0];
  const float* weight = (const float*)d_in[1];
  const float* bias = (const float*)d_in[2];
  float* y = (float*)d_out;

  dim3 grid((G_TOTAL + GC - 1) / GC, T_TOTAL / TC);  // (120, 64)
  dim3 block(256, 1, 1);
  splitlinear_async_kernel<<<grid, block, 0, stream>>>(x, weight, bias, y);
}